// FMNet_54966991454377
// MI455X (gfx1250) — compile-verified
//
#include <hip/hip_runtime.h>
#include <hip/hip_bf16.h>
#include <math.h>

typedef __attribute__((ext_vector_type(2))) float v2f;
typedef __attribute__((ext_vector_type(8))) float v8f;

#define TWO_PI_D    6.28318530717958647692
#define TWO_PI_F    6.2831853071795864769f
#define INV_TWO_PI  0.15915494309189533577f

// One oscillator term, phase tracked in CYCLES (turns):
//   c = {f, s, s*off, w};  pc = phase offset in cycles
//   bell = rsqrt(1 + (tn*s + s*off)^2)
//   u = f*t + pc;  d = u - rint(u) in [-0.5,0.5];  wave = sin/cos(2*pi*d)
// Returns wave*bell; the softmax weight is applied by the WMMA B operand.
template <bool USE_SIN>
__device__ __forceinline__ float term(const float4 c, float tv, float tn, float pc) {
    float arg  = __builtin_fmaf(tn, c.y, c.z);
    float bell = __builtin_amdgcn_rsqf(__builtin_fmaf(arg, arg, 1.0f));
    float u    = __builtin_fmaf(c.x, tv, pc);
    float d    = u - __builtin_rintf(u);      // exact (Sterbenz)
    float r    = d * TWO_PI_F;
    float w    = USE_SIN ? __builtin_amdgcn_sinf(r) : __builtin_amdgcn_cosf(r);
    return w * bell;
}

// ---------------- pre-pass: fold all per-k constants ------------------------
// modtab[k] = {mod_fq, s_m, s_m*off_m, wm}   (frequency in Hz = cycles/sec)
// cartab[k] = {car_fq, s_c, s_c*off_c, wc}
// wmarr/wcarr = contiguous softmax weights (B-operand pairs)
// scal = {sigmoid(phase_offset) [cycles], 1/t_max}
__global__ __launch_bounds__(64) void fm_prep(
    const float* __restrict__ t,   const float* __restrict__ cfq,
    const float* __restrict__ cw,  const float* __restrict__ mfq,
    const float* __restrict__ mw,  const float* __restrict__ po,
    const float* __restrict__ ces, const float* __restrict__ ceo,
    const float* __restrict__ mes, const float* __restrict__ meo,
    float4* __restrict__ modtab, float4* __restrict__ cartab,
    float* __restrict__ wmarr,   float* __restrict__ wcarr,
    float* __restrict__ scal, int T)
{
    __shared__ float sc[64], sm[64];
    const int k = threadIdx.x;   // 64 threads, one per oscillator
    sc[k] = cw[k];
    sm[k] = mw[k];
    __syncthreads();

    float mc = -1e30f, mm = -1e30f;
    for (int i = 0; i < 64; ++i) {
        mc = fmaxf(mc, sc[i]);
        mm = fmaxf(mm, sm[i]);
    }
    float sumc = 0.f, summ = 0.f;
    for (int i = 0; i < 64; ++i) {
        sumc += __expf(sc[i] - mc);
        summ += __expf(sm[i] - mm);
    }
    float wc  = __expf(sc[k] - mc) / sumc;
    float wmv = __expf(sm[k] - mm) / summ;

    // envelope params: s = 2^(sigmoid(slope)*10 - 2), off = tanh(offset)*0.5
    float sg_c = 1.0f / (1.0f + __expf(-ces[k]));
    float s_c  = exp2f(__builtin_fmaf(sg_c, 10.0f, -2.0f));
    float o_c  = tanhf(ceo[k]) * 0.5f;
    float sg_m = 1.0f / (1.0f + __expf(-mes[k]));
    float s_m  = exp2f(__builtin_fmaf(sg_m, 10.0f, -2.0f));
    float o_m  = tanhf(meo[k]) * 0.5f;

    cartab[k] = make_float4(cfq[k], s_c, s_c * o_c, wc);
    modtab[k] = make_float4(mfq[k], s_m, s_m * o_m, wmv);
    wcarr[k]  = wc;
    wmarr[k]  = wmv;

    if (k == 0) {
        scal[0] = 1.0f / (1.0f + __expf(-po[0]));  // phase offset in cycles
        scal[1] = 1.0f / t[T - 1];                 // t increasing => max = t[T-1]
    }
}

// ---------------- main kernel: 1 wave = 16 samples, WMMA K-reduction --------
// A 16x4 f32 layout: lanes 0-15 -> M=lane, K={0,1}; lanes 16-31 -> M=lane-16,
// K={2,3}.  B 4x16 layout mirrors A's K mapping, so lane l's B pair is just
// {w[k0], w[k0+1]} (independent of N) -> one ds_load_b64 per step.
// D[m,n] = sum_k (wave*bell)[m,k] * w[k]; lanes 0/16 hold the 16 row sums.
__global__ __launch_bounds__(256) void fm_main(
    const float*  __restrict__ t,    const float4* __restrict__ mtab,
    const float4* __restrict__ ctab, const float*  __restrict__ wmarr,
    const float*  __restrict__ wcarr, const float* __restrict__ scal,
    float* __restrict__ out, int T)
{
    __shared__ float4 s_mtab[64];
    __shared__ float4 s_ctab[64];
    __shared__ __align__(16) float s_mw[64];
    __shared__ __align__(16) float s_cw[64];
    __shared__ float s_msum[8 * 16];   // per-wave modulator sums

    const int tid = threadIdx.x;
    if (tid < 64)       s_mtab[tid]      = mtab[tid];
    else if (tid < 128) s_ctab[tid - 64] = ctab[tid - 64];
    else if (tid < 192) s_mw[tid - 128]  = wmarr[tid - 128];
    else                s_cw[tid - 192]  = wcarr[tid - 192];
    const float pc0   = scal[0];   // phase offset in cycles
    const float invtm = scal[1];
    __syncthreads();

    const int lane = tid & 31;
    const int wave = tid >> 5;
    const int row  = lane & 15;   // sample within tile
    const int hi   = lane >> 4;   // 0: k-pair {0,1}; 1: k-pair {2,3}
    const int nTiles = (T + 15) >> 4;
    const int gw = blockIdx.x * (blockDim.x >> 5) + wave;
    const int nw = gridDim.x * (blockDim.x >> 5);

    for (int tile = gw; tile < nTiles; tile += nw) {
        int   sidx = tile * 16 + row;
        int   lidx = sidx < T ? sidx : (T - 1);
        float tv   = t[lidx];
        float tn   = __builtin_fmaf(tv, invtm, -0.5f);   // t/t_max - 0.5

        // ---- modulator bank: mod[m] = sum_k cos(2pi(f t + pc0))*bell*w ----
        v8f acc = {0.f, 0.f, 0.f, 0.f, 0.f, 0.f, 0.f, 0.f};
        #pragma unroll
        for (int step = 0; step < 16; ++step) {
            const int p  = step * 2 + hi;   // k-pair index
            const int k0 = p * 2;
            float4 c0 = s_mtab[k0];
            float4 c1 = s_mtab[k0 + 1];
            v2f a, b;
            a.x = term<false>(c0, tv, tn, pc0);
            a.y = term<false>(c1, tv, tn, pc0);
            b = *reinterpret_cast<const v2f*>(&s_mw[k0]);   // {w[k0], w[k0+1]}
            acc = __builtin_amdgcn_wmma_f32_16x16x4_f32(
                false, a, false, b, (short)0, acc, false, false);
        }

        // route mod sums (lanes 0/16 own them) to all lanes via LDS
        if (lane == 0 || lane == 16) {
            float* dst = &s_msum[wave * 16 + hi * 8];
            #pragma unroll
            for (int r = 0; r < 8; ++r) dst[r] = acc[r];
        }
        asm volatile("s_wait_dscnt 0" ::: "memory");  // LDS in-order per wave
        float modv = s_msum[wave * 16 + row];

        // ---- carrier bank: sin(2pi(f t + pc0) + mod)*bell*w ----
        float pcc = __builtin_fmaf(modv, INV_TWO_PI, pc0);  // cycles
        v8f acc2 = {0.f, 0.f, 0.f, 0.f, 0.f, 0.f, 0.f, 0.f};
        #pragma unroll
        for (int step = 0; step < 16; ++step) {
            const int p  = step * 2 + hi;
            const int k0 = p * 2;
            float4 c0 = s_ctab[k0];
            float4 c1 = s_ctab[k0 + 1];
            v2f a, b;
            a.x = term<true>(c0, tv, tn, pcc);
            a.y = term<true>(c1, tv, tn, pcc);
            b = *reinterpret_cast<const v2f*>(&s_cw[k0]);
            acc2 = __builtin_amdgcn_wmma_f32_16x16x4_f32(
                false, a, false, b, (short)0, acc2, false, false);
        }

        // lanes 0/16 hold carrier[tile*16 + 0..7] / [8..15] in C VGPRs
        const int obase = tile * 16 + hi * 8;
        if (lane == 0 || lane == 16) {
            if (obase + 8 <= T) {
                float4 lo4 = make_float4(acc2[0], acc2[1], acc2[2], acc2[3]);
                float4 hi4 = make_float4(acc2[4], acc2[5], acc2[6], acc2[7]);
                float4* ptr = reinterpret_cast<float4*>(out + obase);
                ptr[0] = lo4;
                ptr[1] = hi4;
            } else {
                #pragma unroll
                for (int r = 0; r < 8; ++r)
                    if (obase + r < T) out[obase + r] = acc2[r];
            }
        }
    }
}

// ---------------------------------------------------------------------------
extern "C" void kernel_launch(void* const* d_in, const int* in_sizes, int n_in,
                              void* d_out, int out_size, void* d_ws, size_t ws_size,
                              hipStream_t stream)
{
    const float* t   = (const float*)d_in[0];
    const float* cfq = (const float*)d_in[1];
    const float* cw  = (const float*)d_in[2];
    const float* mfq = (const float*)d_in[3];
    const float* mw  = (const float*)d_in[4];
    const float* po  = (const float*)d_in[5];
    const float* ces = (const float*)d_in[6];
    const float* ceo = (const float*)d_in[7];
    const float* mes = (const float*)d_in[8];
    const float* meo = (const float*)d_in[9];
    const int T = in_sizes[0];

    char* ws = (char*)d_ws;
    float4* mtab  = (float4*)(ws);            // 64 * 16B
    float4* ctab  = (float4*)(ws + 1024);     // 64 * 16B
    float*  wmarr = (float*) (ws + 2048);     // 64 * 4B
    float*  wcarr = (float*) (ws + 2304);     // 64 * 4B
    float*  scal  = (float*) (ws + 2560);     // 2 floats

    fm_prep<<<1, 64, 0, stream>>>(t, cfq, cw, mfq, mw, po, ces, ceo, mes, meo,
                                  mtab, ctab, wmarr, wcarr, scal, T);

    const int nTiles = (T + 15) >> 4;
    const int wavesPerBlock = 8;         // 256 threads, wave32
    int blocks = (nTiles + wavesPerBlock - 1) / wavesPerBlock;
    if (blocks > 2048) blocks = 2048;
    if (blocks < 1) blocks = 1;

    fm_main<<<blocks, 256, 0, stream>>>(t, mtab, ctab, wmarr, wcarr, scal,
                                        (float*)d_out, T);
}